// ServerPixelContrastLoss_65575560675708
// MI455X (gfx1250) — compile-verified
//
#include <hip/hip_runtime.h>
#include <hip/hip_bf16.h>
#include <math.h>

// ---------------------------------------------------------------------------
// Supervised pixel-contrast loss on gfx1250 (MI455X), fp32 end-to-end via
// V_WMMA_F32_16X16X4_F32.  M=16384 anchor rows, N=4864 contrast cols, D=256.
// Each wave owns 16 rows x all columns -> fully deterministic reductions.
// ---------------------------------------------------------------------------

typedef __attribute__((ext_vector_type(2))) float v2f;
typedef __attribute__((ext_vector_type(8))) float v8f;

#define M_ROWS 16384
#define N_COLS 4864
#define D_DIM  256
#define ANUM   64
#define NVIEW  256
#define TEMP   0.07f
#define EPSV   1e-10f

// ---- kernel 0: per-row inverse-norm / temperature scale -------------------
__global__ void __launch_bounds__(256) rownorm_kernel(const float* __restrict__ Xa,
                                                      float* __restrict__ invT) {
    int wid  = (blockIdx.x * blockDim.x + threadIdx.x) >> 5;   // 0..16383
    int lane = threadIdx.x & 31;
    int a = wid & (ANUM - 1);
    int v = wid >> 6;
    const float* base = Xa + (size_t)a * (NVIEW * D_DIM) + (size_t)v * D_DIM + lane * 8;
    float4 f0 = *(const float4*)(base);
    float4 f1 = *(const float4*)(base + 4);
    float ss = f0.x*f0.x + f0.y*f0.y + f0.z*f0.z + f0.w*f0.w
             + f1.x*f1.x + f1.y*f1.y + f1.z*f1.z + f1.w*f1.w;
    #pragma unroll
    for (int off = 16; off > 0; off >>= 1) ss += __shfl_xor(ss, off, 32);
    if (lane == 0) invT[wid] = 1.0f / (fmaxf(sqrtf(ss), 1e-12f) * TEMP);
}

// ---- kernel 1: positives-per-anchor-class count ---------------------------
__global__ void poscount_kernel(const int* __restrict__ ya,
                                const int* __restrict__ yc,
                                float* __restrict__ pc) {
    int a = threadIdx.x;            // 64 threads
    int cls = ya[a];
    int cnt = 0;
    for (int j = 0; j < N_COLS; ++j) cnt += (yc[j] == cls) ? 1 : 0;
    pc[a] = (float)cnt;
}

// ---- in-register 16x64 tile GEMM over full K=256 via f32 WMMA -------------
__device__ __forceinline__ void gemm16x64(const float* __restrict__ aPtr,
                                          const float* __restrict__ b0,
                                          const float* __restrict__ b1,
                                          const float* __restrict__ b2,
                                          const float* __restrict__ b3,
                                          v8f& c0, v8f& c1, v8f& c2, v8f& c3) {
    v8f z = {};
    c0 = z; c1 = z; c2 = z; c3 = z;
    #pragma unroll 4
    for (int k = 0; k < D_DIM; k += 4) {
        v2f af  = *(const v2f*)(aPtr + k);
        v2f bf0 = *(const v2f*)(b0 + k);
        v2f bf1 = *(const v2f*)(b1 + k);
        v2f bf2 = *(const v2f*)(b2 + k);
        v2f bf3 = *(const v2f*)(b3 + k);
        c0 = __builtin_amdgcn_wmma_f32_16x16x4_f32(false, af, false, bf0, (short)0, c0, false, false);
        c1 = __builtin_amdgcn_wmma_f32_16x16x4_f32(false, af, false, bf1, (short)0, c1, false, false);
        c2 = __builtin_amdgcn_wmma_f32_16x16x4_f32(false, af, false, bf2, (short)0, c2, false, false);
        c3 = __builtin_amdgcn_wmma_f32_16x16x4_f32(false, af, false, bf3, (short)0, c3, false, false);
    }
}

// online max / neg-exp-sum update for one tile column-group
#define ONLINE_UPD(T, YCV)                                                     \
    { float l  = (T)[vg] * scale[vg];                                          \
      float nm = fmaxf(mrow[vg], l);                                           \
      Srow[vg] = Srow[vg] * __expf(mrow[vg] - nm)                              \
               + ((yrow[vg] == (YCV)) ? 0.0f : __expf(l - nm));                \
      mrow[vg] = nm; }

// positive-term accumulation for one tile column-group
#define POS_UPD(T, YCV)                                                        \
    { float l    = (T)[vg] * scale[vg] - mrow[vg];                             \
      float term = l - __logf(__expf(l) + Sref[vg] + EPSV);                    \
      accp[vg]  += (yrow[vg] == (YCV)) ? term : 0.0f; }

// ---- kernel 2: main fused GEMM + masked-softmax loss ----------------------
__global__ void __launch_bounds__(256) contrast_main_kernel(
        const float* __restrict__ Xa, const int* __restrict__ yA,
        const float* __restrict__ Xc, const int* __restrict__ yC,
        const float* __restrict__ invT, const float* __restrict__ posCnt,
        float* __restrict__ partials) {
    const int lane    = threadIdx.x & 31;
    const int waveId  = (blockIdx.x * blockDim.x + threadIdx.x) >> 5;  // 0..1023
    const int rowBase = waveId * 16;
    const int half    = lane >> 4;     // K / row-half selector
    const int l16     = lane & 15;

    // A fragment source: lane l16 supplies row rowBase+l16, K pair (2*half)
    int rowA = rowBase + l16;
    const float* aPtr = Xa + (size_t)(rowA & (ANUM - 1)) * (NVIEW * D_DIM)
                           + (size_t)(rowA >> 6) * D_DIM + 2 * half;

    // per-lane C-element rows: rowBase + 8*half + vg  (vg = 0..7)
    float scale[8]; int yrow[8]; float npos[8];
    #pragma unroll
    for (int vg = 0; vg < 8; ++vg) {
        int rr    = rowBase + 8 * half + vg;
        scale[vg] = invT[rr];
        yrow[vg]  = yA[rr & (ANUM - 1)];
        npos[vg]  = posCnt[rr & (ANUM - 1)];
    }

    float mrow[8], Srow[8];
    #pragma unroll
    for (int vg = 0; vg < 8; ++vg) { mrow[vg] = -INFINITY; Srow[vg] = 0.0f; }

    // -------- pass 1: online row-max + negative exp-sum --------------------
    for (int cb = 0; cb < N_COLS; cb += 64) {
        int c0 = cb + l16;
        const float* b0 = Xc + (size_t)c0 * D_DIM + 2 * half;
        const float* b1 = b0 + 16 * D_DIM;
        const float* b2 = b0 + 32 * D_DIM;
        const float* b3 = b0 + 48 * D_DIM;
        int yc0 = yC[c0], yc1 = yC[c0 + 16], yc2 = yC[c0 + 32], yc3 = yC[c0 + 48];
        v8f t0, t1, t2, t3;
        gemm16x64(aPtr, b0, b1, b2, b3, t0, t1, t2, t3);
        #pragma unroll
        for (int vg = 0; vg < 8; ++vg) {
            ONLINE_UPD(t0, yc0)
            ONLINE_UPD(t1, yc1)
            ONLINE_UPD(t2, yc2)
            ONLINE_UPD(t3, yc3)
        }
    }

    // merge (m,S) across the 16 lanes holding each row (flash-style combine)
    #pragma unroll
    for (int vg = 0; vg < 8; ++vg) {
        #pragma unroll
        for (int off = 1; off < 16; off <<= 1) {
            float mo = __shfl_xor(mrow[vg], off, 16);
            float so = __shfl_xor(Srow[vg], off, 16);
            float nm = fmaxf(mrow[vg], mo);
            Srow[vg] = Srow[vg] * __expf(mrow[vg] - nm) + so * __expf(mo - nm);
            mrow[vg] = nm;
        }
    }

    float Sref[8], accp[8];
    #pragma unroll
    for (int vg = 0; vg < 8; ++vg) { Sref[vg] = Srow[vg] + npos[vg]; accp[vg] = 0.0f; }

    // -------- pass 2: positive log-prob terms ------------------------------
    for (int cb = 0; cb < N_COLS; cb += 64) {
        int c0 = cb + l16;
        const float* b0 = Xc + (size_t)c0 * D_DIM + 2 * half;
        const float* b1 = b0 + 16 * D_DIM;
        const float* b2 = b0 + 32 * D_DIM;
        const float* b3 = b0 + 48 * D_DIM;
        int yc0 = yC[c0], yc1 = yC[c0 + 16], yc2 = yC[c0 + 32], yc3 = yC[c0 + 48];
        v8f t0, t1, t2, t3;
        gemm16x64(aPtr, b0, b1, b2, b3, t0, t1, t2, t3);
        #pragma unroll
        for (int vg = 0; vg < 8; ++vg) {
            POS_UPD(t0, yc0)
            POS_UPD(t1, yc1)
            POS_UPD(t2, yc2)
            POS_UPD(t3, yc3)
        }
    }

    // sum positive terms across the 16 lanes of each row-half
    #pragma unroll
    for (int vg = 0; vg < 8; ++vg) {
        #pragma unroll
        for (int off = 1; off < 16; off <<= 1)
            accp[vg] += __shfl_xor(accp[vg], off, 16);
    }

    // per-row: add closed-form negative-column term, divide by n_pos, negate
    float part = 0.0f;
    #pragma unroll
    for (int vg = 0; vg < 8; ++vg) {
        float tot = accp[vg]
                  - ((float)N_COLS - npos[vg]) * __logf(1.0f + Sref[vg] + EPSV);
        part += -(tot / npos[vg]);
    }
    part += __shfl_xor(part, 16, 32);   // combine the two row-halves
    if (lane == 0) partials[waveId] = part;
}

// ---- kernel 3: deterministic final reduction of 1024 wave partials --------
__global__ void __launch_bounds__(256) final_reduce_kernel(const float* __restrict__ partials,
                                                           float* __restrict__ out) {
    __shared__ float s[256];
    int t = threadIdx.x;
    s[t] = partials[t] + partials[t + 256] + partials[t + 512] + partials[t + 768];
    __syncthreads();
    for (int off = 128; off > 0; off >>= 1) {
        if (t < off) s[t] += s[t + off];
        __syncthreads();
    }
    if (t == 0) out[0] = s[0] * (1.0f / (float)M_ROWS);
}

extern "C" void kernel_launch(void* const* d_in, const int* in_sizes, int n_in,
                              void* d_out, int out_size, void* d_ws, size_t ws_size,
                              hipStream_t stream) {
    const float* Xa = (const float*)d_in[0];   // (64, 256, 256) fp32
    const int*   ya = (const int*)  d_in[1];   // (64,) int32
    const float* Xc = (const float*)d_in[2];   // (4864, 256) fp32
    const int*   yc = (const int*)  d_in[3];   // (4864,) int32

    float* ws       = (float*)d_ws;
    float* invT     = ws;                      // 16384 floats
    float* posCnt   = ws + M_ROWS;             // 64 floats
    float* partials = ws + M_ROWS + ANUM;      // 1024 floats

    rownorm_kernel      <<<M_ROWS / 8, 256, 0, stream>>>(Xa, invT);
    poscount_kernel     <<<1, ANUM, 0, stream>>>(ya, yc, posCnt);
    contrast_main_kernel<<<(M_ROWS / 16) / 8, 256, 0, stream>>>(Xa, ya, Xc, yc,
                                                               invT, posCnt, partials);
    final_reduce_kernel <<<1, 256, 0, stream>>>(partials, (float*)d_out);
}